// RiskGNN_5574867550344
// MI455X (gfx1250) — compile-verified
//
#include <hip/hip_runtime.h>
#include <hip/hip_bf16.h>

// ---------------------------------------------------------------------------
// GATv2 x2 + linear head for MI455X (gfx1250, wave32, WMMA).
// - Node transforms: V_WMMA_F32_16X16X4_F32 (fp32-exact vs reference).
//   A tiles staged global->LDS with GLOBAL_LOAD_ASYNC_TO_LDS_B128 (ASYNCcnt),
//   shared by all waves of a block; B (weights) double-buffered from L0/L2.
// - Edge passes: one wave32 per edge, lane == channel (HID == 32 == wave32).
//   Segment softmax: native global_atomic_max_num_f32 + global_atomic_add_f32.
//   xl/xr (51 MB) stay resident in the 192 MB L2 -> gathers are L2-bound.
// ---------------------------------------------------------------------------

typedef float v2f  __attribute__((ext_vector_type(2)));
typedef float v8f  __attribute__((ext_vector_type(8)));

#define WAVE 32
#define NEG_SLOPE 0.2f
#define K_DIM 128           // inner dim of both GEMM layers
#define LDW 132             // LDS row stride (pad: 132 % 64 == 4 -> conflict-free b64)

// ------------------------------- utilities ---------------------------------

__global__ void k_fill(float* __restrict__ p, float v, long long n) {
    long long t = (long long)blockIdx.x * blockDim.x + threadIdx.x;
    if (t < n) p[t] = v;
}

// gfx1250 native float atomic max, no return (segment max doesn't need old val)
__device__ __forceinline__ void atomic_max_f32(float* addr, float val) {
    asm volatile("global_atomic_max_num_f32 %0, %1, off scope:SCOPE_DEV"
                 :: "v"(addr), "v"(val) : "memory");
}

__device__ __forceinline__ float wave_sum(float s) {
#pragma unroll
    for (int off = 16; off; off >>= 1) s += __shfl_xor(s, off, WAVE);
    return s;
}

// ------------------------------ WMMA GEMM ----------------------------------
// C[M,NC] = A[M,K_DIM] * W[K_DIM,NC] + bias[NC], NC = 16*TILES_N.
// Block = 8 wave32. Waves with the same tile_m share one A tile (16x128,
// contiguous 8KB in global) staged into LDS via async b128 copies
// (one instruction per 512B row). WMMA fragments per ISA 7.12.2:
//   A (16x4, 2 VGPR): lane<16 -> M=lane, K={k,k+1}; lane>=16 -> K={k+2,k+3}
//   B (4x16, 2 VGPR): lane<16 -> N=lane, rows {k,k+1}; lane>=16 -> {k+2,k+3}
//   C/D (8 VGPR): vgpr r -> row r (lanes 0-15) / row r+8 (lanes 16-31)

template <int TILES_N>
__global__ void k_gemm_wmma(const float* __restrict__ A,
                            const float* __restrict__ W,
                            const float* __restrict__ bias,
                            float* __restrict__ C,
                            int M) {
    constexpr int WPB     = 8;                  // waves per block
    constexpr int ROWS_PB = WPB / TILES_N;      // tile_m rows per block
    constexpr int NC      = TILES_N * 16;
    __shared__ float sA[ROWS_PB * 16 * LDW];

    const int lane   = threadIdx.x & (WAVE - 1);
    const int wib    = threadIdx.x >> 5;        // wave index in block
    const int tilesM = M >> 4;

    // ---- async-stage A rows (512B each, contiguous) into LDS ----
    const int rowBase = blockIdx.x * (ROWS_PB * 16);
    for (int r = wib; r < ROWS_PB * 16; r += WPB) {
        if (rowBase + r < M) {
            const float* g = A + (size_t)(rowBase + r) * K_DIM + lane * 4;
            unsigned     l = (unsigned)(size_t)(&sA[r * LDW + lane * 4]);
            asm volatile("global_load_async_to_lds_b128 %0, %1, off"
                         :: "v"(l), "v"(g) : "memory");
        }
    }
    asm volatile("s_wait_asynccnt 0x0" ::: "memory");
    __syncthreads();

    // ---- per-wave 16x16 tile ----
    const int tm_local = wib / TILES_N;
    const int tile_m   = blockIdx.x * ROWS_PB + tm_local;
    const int tile_n   = wib - tm_local * TILES_N;
    if (tile_m >= tilesM) return;

    const int l15  = lane & 15;
    const int koff = (lane >> 4) << 1;                 // 0 or 2
    const int n    = tile_n * 16 + l15;                // output column
    const float* la = &sA[(tm_local * 16 + l15) * LDW + koff];
    const float* wc = W + n;                           // weight column

    v8f acc = {};
    // software pipeline: preload next K-step's fragments before the WMMA
    v2f   a  = *(const v2f*)(la);                      // ds_load_b64
    float b0 = wc[(size_t)koff * NC];
    float b1 = wc[(size_t)(koff + 1) * NC];
#pragma unroll
    for (int k = 0; k < K_DIM; k += 4) {
        v2f an = a; float b0n = b0, b1n = b1;
        if (k + 4 < K_DIM) {
            an  = *(const v2f*)(la + k + 4);
            b0n = wc[(size_t)(k + 4 + koff) * NC];
            b1n = wc[(size_t)(k + 5 + koff) * NC];
        }
        v2f b; b.x = b0; b.y = b1;
        acc = __builtin_amdgcn_wmma_f32_16x16x4_f32(
                /*neg_a=*/false, a, /*neg_b=*/false, b,
                /*c_mod=*/(short)0, acc, /*reuse_a=*/false, /*reuse_b=*/false);
        a = an; b0 = b0n; b1 = b1n;
    }

    const float bn = bias[n];
    const int   rb = tile_m * 16 + ((lane >> 4) << 3); // +8 for hi half-wave
#pragma unroll
    for (int r = 0; r < 8; ++r)
        C[(size_t)(rb + r) * NC + n] = acc[r] + bn;
}

// ---------------------------- edge kernels ---------------------------------
// Edges 0..E-1 from edge_index, edges E..E+Nn-1 are self loops.

template <int H>
__global__ void k_edge_score(const float* __restrict__ xl,
                             const float* __restrict__ xr,
                             const float* __restrict__ att,
                             const int* __restrict__ ei,
                             int E, int Nn,
                             float* __restrict__ score,
                             float* __restrict__ mseg) {
    const int lane = threadIdx.x & (WAVE - 1);
    const int e    = (blockIdx.x * (blockDim.x >> 5)) + (threadIdx.x >> 5);
    const int ET   = E + Nn;
    if (e >= ET) return;
    int src, dst;
    if (e < E) { src = ei[e]; dst = ei[E + e]; }
    else       { src = dst = e - E; }

    const int HC = H * 32;
#pragma unroll
    for (int h = 0; h < H; ++h) {
        float v = xl[(size_t)src * HC + h * 32 + lane] +
                  xr[(size_t)dst * HC + h * 32 + lane];
        v = v > 0.0f ? v : NEG_SLOPE * v;                 // LeakyReLU
        float s = wave_sum(v * att[h * 32 + lane]);       // a^T . e
        if (lane == 0) {
            score[(size_t)e * H + h] = s;
            atomic_max_f32(&mseg[(size_t)dst * H + h], s);
        }
    }
}

template <int H>
__global__ void k_edge_exp(const int* __restrict__ ei, int E, int Nn,
                           float* __restrict__ score,          // in: s, out: exp(s-m)
                           const float* __restrict__ mseg,
                           float* __restrict__ denom) {
    long long t = (long long)blockIdx.x * blockDim.x + threadIdx.x;
    long long total = (long long)(E + Nn) * H;
    if (t >= total) return;
    int e = (int)(t / H);
    int h = (int)(t - (long long)e * H);
    int dst = (e < E) ? ei[E + e] : (e - E);
    float ex = __expf(score[t] - mseg[(size_t)dst * H + h]);
    score[t] = ex;
    atomicAdd(&denom[(size_t)dst * H + h], ex);           // global_atomic_add_f32
}

template <int H>
__global__ void k_edge_agg(const float* __restrict__ xl,
                           const int* __restrict__ ei, int E, int Nn,
                           const float* __restrict__ ex,
                           const float* __restrict__ denom,
                           float* __restrict__ agg) {
    const int lane = threadIdx.x & (WAVE - 1);
    const int e    = (blockIdx.x * (blockDim.x >> 5)) + (threadIdx.x >> 5);
    const int ET   = E + Nn;
    if (e >= ET) return;
    int src, dst;
    if (e < E) { src = ei[e]; dst = ei[E + e]; }
    else       { src = dst = e - E; }

    const int HC = H * 32;
#pragma unroll
    for (int h = 0; h < H; ++h) {
        float alpha = ex[(size_t)e * H + h] / denom[(size_t)dst * H + h];
        float v     = alpha * xl[(size_t)src * HC + h * 32 + lane];
        atomicAdd(&agg[(size_t)dst * HC + h * 32 + lane], v);
    }
}

__global__ void k_bias_elu(float* __restrict__ a, const float* __restrict__ bias,
                           long long n, int cols) {
    long long t = (long long)blockIdx.x * blockDim.x + threadIdx.x;
    if (t >= n) return;
    float v = a[t] + bias[t % cols];
    a[t] = v > 0.0f ? v : (__expf(v) - 1.0f);             // ELU
}

// final: out[n,:2] = h[n,:32] @ Wlin[32,2] + blin   (wave32 per node, lane=c)
__global__ void k_final(const float* __restrict__ h,
                        const float* __restrict__ Wlin,
                        const float* __restrict__ blin,
                        float* __restrict__ out, int Nn) {
    const int lane = threadIdx.x & (WAVE - 1);
    const int nidx = (blockIdx.x * (blockDim.x >> 5)) + (threadIdx.x >> 5);
    if (nidx >= Nn) return;
    float v  = h[(size_t)nidx * 32 + lane];
    float p0 = wave_sum(v * Wlin[lane * 2 + 0]);
    float p1 = wave_sum(v * Wlin[lane * 2 + 1]);
    if (lane == 0) {
        out[(size_t)nidx * 2 + 0] = p0 + blin[0];
        out[(size_t)nidx * 2 + 1] = p1 + blin[1];
    }
}

// ------------------------------- launcher ----------------------------------

static inline int cdiv(long long a, long long b) { return (int)((a + b - 1) / b); }

extern "C" void kernel_launch(void* const* d_in, const int* in_sizes, int n_in,
                              void* d_out, int out_size, void* d_ws, size_t ws_size,
                              hipStream_t stream) {
    const float* x     = (const float*)d_in[0];
    const int*   ei    = (const int*)  d_in[1];
    const float* W1l   = (const float*)d_in[2];
    const float* b1l   = (const float*)d_in[3];
    const float* W1r   = (const float*)d_in[4];
    const float* b1r   = (const float*)d_in[5];
    const float* att1  = (const float*)d_in[6];
    const float* bias1 = (const float*)d_in[7];
    const float* W2l   = (const float*)d_in[8];
    const float* b2l   = (const float*)d_in[9];
    const float* W2r   = (const float*)d_in[10];
    const float* b2r   = (const float*)d_in[11];
    const float* att2  = (const float*)d_in[12];
    const float* bias2 = (const float*)d_in[13];
    const float* Wlin  = (const float*)d_in[14];
    const float* blin  = (const float*)d_in[15];
    float* out = (float*)d_out;

    const int IN   = 128;
    const int HID  = 32;
    const int HDS  = 4;
    const int N    = in_sizes[0] / IN;     // 100000
    const int E    = in_sizes[1] / 2;      // 1600000
    const int ET   = E + N;                // with self loops
    const int HC1  = HDS * HID;            // 128

    // ---- workspace carve (floats) ----
    float* ws = (float*)d_ws;
    size_t o = 0;
    float* xl1  = ws + o; o += (size_t)N * HC1;
    float* xr1  = ws + o; o += (size_t)N * HC1;
    float* agg1 = ws + o; o += (size_t)N * HC1;
    float* xl2  = ws + o; o += (size_t)N * HID;
    float* xr2  = ws + o; o += (size_t)N * HID;
    float* agg2 = ws + o; o += (size_t)N * HID;
    float* sc1  = ws + o; o += (size_t)ET * HDS;
    float* sc2  = ws + o; o += (size_t)ET;
    float* m1   = ws + o; o += (size_t)N * HDS;
    float* dn1  = ws + o; o += (size_t)N * HDS;
    float* m2   = ws + o; o += (size_t)N;
    float* dn2  = ws + o; o += (size_t)N;

    const int TB  = 256;                   // 8 wave32 per block
    const int WPB = TB / WAVE;
    const int tilesM = N / 16;             // 6250 (exact)

    // ---- init (atomic targets must be re-initialized every call) ----
    k_fill<<<cdiv((long long)N*HC1, TB), TB, 0, stream>>>(agg1, 0.0f, (long long)N*HC1);
    k_fill<<<cdiv((long long)N*HID, TB), TB, 0, stream>>>(agg2, 0.0f, (long long)N*HID);
    k_fill<<<cdiv((long long)N*HDS, TB), TB, 0, stream>>>(dn1,  0.0f, (long long)N*HDS);
    k_fill<<<cdiv((long long)N,     TB), TB, 0, stream>>>(dn2,  0.0f, (long long)N);
    k_fill<<<cdiv((long long)N*HDS, TB), TB, 0, stream>>>(m1, -1e30f, (long long)N*HDS);
    k_fill<<<cdiv((long long)N,     TB), TB, 0, stream>>>(m2, -1e30f, (long long)N);

    // ---- layer 1: node transforms (WMMA fp32, async-LDS staged A) ----
    // TILES_N=8 -> one block row per block, A tile shared by all 8 waves
    k_gemm_wmma<8><<<tilesM, TB, 0, stream>>>(x, W1l, b1l, xl1, N);
    k_gemm_wmma<8><<<tilesM, TB, 0, stream>>>(x, W1r, b1r, xr1, N);

    // ---- layer 1: edge softmax + aggregate ----
    k_edge_score<HDS><<<cdiv(ET, WPB), TB, 0, stream>>>(xl1, xr1, att1, ei, E, N, sc1, m1);
    k_edge_exp<HDS><<<cdiv((long long)ET*HDS, TB), TB, 0, stream>>>(ei, E, N, sc1, m1, dn1);
    k_edge_agg<HDS><<<cdiv(ET, WPB), TB, 0, stream>>>(xl1, ei, E, N, sc1, dn1, agg1);
    k_bias_elu<<<cdiv((long long)N*HC1, TB), TB, 0, stream>>>(agg1, bias1, (long long)N*HC1, HC1);

    // ---- layer 2: node transforms (WMMA fp32), K=128 -> 32 ----
    // TILES_N=2 -> 4 tile_m rows per block (guarded tail)
    k_gemm_wmma<2><<<cdiv(tilesM, 4), TB, 0, stream>>>(agg1, W2l, b2l, xl2, N);
    k_gemm_wmma<2><<<cdiv(tilesM, 4), TB, 0, stream>>>(agg1, W2r, b2r, xr2, N);

    // ---- layer 2: edge softmax + aggregate (1 head) ----
    k_edge_score<1><<<cdiv(ET, WPB), TB, 0, stream>>>(xl2, xr2, att2, ei, E, N, sc2, m2);
    k_edge_exp<1><<<cdiv((long long)ET, TB), TB, 0, stream>>>(ei, E, N, sc2, m2, dn2);
    k_edge_agg<1><<<cdiv(ET, WPB), TB, 0, stream>>>(xl2, ei, E, N, sc2, dn2, agg2);
    k_bias_elu<<<cdiv((long long)N*HID, TB), TB, 0, stream>>>(agg2, bias2, (long long)N*HID, HID);

    // ---- head: [N,32] @ [32,2] + blin ----
    k_final<<<cdiv(N, WPB), TB, 0, stream>>>(agg2, Wlin, blin, out, N);
}